// PETNet_27693949124724
// MI455X (gfx1250) — compile-verified
//
#include <hip/hip_runtime.h>
#include <math.h>

// ---------------------------------------------------------------------------
// SNN: T=25, B=4096, NIN=784, NH=1000, NOUT=10, THRESH=1.0
// Layer1: fp32 WMMA 16x16x4 GEMM fused with LIF update (state in d_ws),
//         double-buffered LDS staging + double-buffered fragment registers
//         with sched_barrier-pinned pipelining.
// Layer2: tiny GEMM (N=10) via wave-per-row dot + shuffle reduce + LIF
// ---------------------------------------------------------------------------

#define T_STEPS 25
#define BATCH   4096
#define NIN     784
#define NH      1000
#define NOUT    10

typedef __attribute__((ext_vector_type(2))) float v2f;
typedef __attribute__((ext_vector_type(8))) float v8f;

// LDS tile row pitch (floats). 20 keeps float4 stores 16B aligned (20*4=80)
// and gives stride-20 fragment reads (gcd(20,64)=4 -> no bank conflicts).
#define LPAD 20

// ---------------------------------------------------------------------------
// Kernel A: one time step of layer 1.
//   cur1  = x_t @ W1^T           (WMMA f32 16x16x4, 128x128 tile / block)
//   mem1' = beta1*mem1 + cur1 + b1 - (mem1 > 1)
//   spk1  = (mem1' > 1)
// grid = (ceil(NH/128)=8, BATCH/128=32), block = 256 (8 waves)
// ---------------------------------------------------------------------------
__global__ __launch_bounds__(256) void snn_layer1_wmma(
    const float* __restrict__ x,     // [T,B,NIN], we index step t
    const float* __restrict__ W1,    // [NH, NIN] row-major
    const float* __restrict__ b1,    // [NH]
    const float* __restrict__ beta1p,// [1]
    float* __restrict__ mem1,        // [B, NH] persistent state (ws)
    float* __restrict__ spk1,        // [B, NH] (ws)
    int t)
{
    __shared__ float As[2][128 * LPAD]; // A tile: [m 0..127][k 0..15]
    __shared__ float Bs[2][128 * LPAD]; // B tile: [n 0..127][k 0..15]

    const int tid      = threadIdx.x;
    const int lane     = tid & 31;
    const int wave     = tid >> 5;          // 0..7
    const int half     = lane >> 4;         // 0 / 1
    const int lr       = lane & 15;         // 0..15
    const int rowBase  = blockIdx.y * 128;  // batch rows
    const int colBase  = blockIdx.x * 128;  // hidden units
    const int m0       = wave * 16;         // wave's 16-row strip in tile

    const float* X = x + (size_t)t * BATCH * NIN;

    // per-thread staging coordinates (2 float4 each for A and B tiles)
    int srow[2], skq[2], sgn[2];
#pragma unroll
    for (int i = 0; i < 2; ++i) {
        int idx = tid + i * 256;   // 0..511
        srow[i] = idx >> 2;        // 0..127
        skq[i]  = (idx & 3) << 2;  // 0,4,8,12
        sgn[i]  = colBase + srow[i];
    }

    v8f acc[8];
#pragma unroll
    for (int tn = 0; tn < 8; ++tn) acc[tn] = (v8f)0.0f;

    float4 aReg[2], bReg[2];

    // ---- prologue: fetch tile 0 and stage into LDS buffer 0 ----
#pragma unroll
    for (int i = 0; i < 2; ++i) {
        aReg[i] = *(const float4*)(X + (size_t)(rowBase + srow[i]) * NIN + skq[i]);
        bReg[i] = make_float4(0.f, 0.f, 0.f, 0.f);
        if (sgn[i] < NH)
            bReg[i] = *(const float4*)(W1 + (size_t)sgn[i] * NIN + skq[i]);
        *(float4*)(&As[0][srow[i] * LPAD + skq[i]]) = aReg[i];
        *(float4*)(&Bs[0][srow[i] * LPAD + skq[i]]) = bReg[i];
    }

    const int NKT = NIN / 16; // 49
    for (int it = 0; it < NKT; ++it) {
        const int buf   = it & 1;
        const int k0n   = (it + 1) * 16;
        const bool more = (it + 1 < NKT);

        // ---- prefetch NEXT tile from global into registers (overlaps WMMA) --
        if (more) {
#pragma unroll
            for (int i = 0; i < 2; ++i) {
                aReg[i] = *(const float4*)(X + (size_t)(rowBase + srow[i]) * NIN + k0n + skq[i]);
                bReg[i] = make_float4(0.f, 0.f, 0.f, 0.f);
                if (sgn[i] < NH)
                    bReg[i] = *(const float4*)(W1 + (size_t)sgn[i] * NIN + k0n + skq[i]);
            }
        }

        __syncthreads();  // LDS buffer `buf` fully staged, prior readers done

        // ---- compute current tile: 4 k-substeps x 8 n-tiles of WMMA,
        //      software-pipelined fragment loads (regs double-buffered) ----
        const float* Ab   = As[buf];
        const float* Bb   = Bs[buf];
        const int    aoff = (m0 + lr) * LPAD + half * 2;
        const int    boff = lr * LPAD + half * 2;

        v2f aF[2];
        v2f bF[2][8];
        aF[0] = *(const v2f*)(&Ab[aoff]);
#pragma unroll
        for (int tn = 0; tn < 8; ++tn)
            bF[0][tn] = *(const v2f*)(&Bb[boff + tn * 16 * LPAD]);

#pragma unroll
        for (int s = 0; s < 4; ++s) {
            const int cur = s & 1;
            const int nxt = cur ^ 1;
            if (s < 3) {                     // load substep s+1 into alt regs
                const int kk = (s + 1) * 4;
                aF[nxt] = *(const v2f*)(&Ab[aoff + kk]);
#pragma unroll
                for (int tn = 0; tn < 8; ++tn)
                    bF[nxt][tn] = *(const v2f*)(&Bb[boff + tn * 16 * LPAD + kk]);
            }
#pragma unroll
            for (int tn = 0; tn < 8; ++tn)
                acc[tn] = __builtin_amdgcn_wmma_f32_16x16x4_f32(
                    false, aF[cur], false, bF[cur][tn], (short)0, acc[tn],
                    false, false);
            // region boundary: loads(s+1) must stay with wmmas(s), so their
            // dscnt wait is covered by 8 WMMAs of latency.
            __builtin_amdgcn_sched_barrier(0);
        }

        // ---- stage next tile into the other LDS buffer ----
        if (more) {
#pragma unroll
            for (int i = 0; i < 2; ++i) {
                *(float4*)(&As[buf ^ 1][srow[i] * LPAD + skq[i]]) = aReg[i];
                *(float4*)(&Bs[buf ^ 1][srow[i] * LPAD + skq[i]]) = bReg[i];
            }
        }
    }

    // ---- epilogue: bias + LIF. C/D layout: VGPR j <-> M = j (lanes<16) / j+8 --
    const float bt1 = fminf(fmaxf(beta1p[0], 0.0f), 1.0f);
    const int rbase = rowBase + m0 + half * 8;
#pragma unroll
    for (int tn = 0; tn < 8; ++tn) {
        int c = colBase + tn * 16 + lr;
        if (c < NH) {
            float bias = b1[c];
#pragma unroll
            for (int j = 0; j < 8; ++j) {
                size_t idx  = (size_t)(rbase + j) * NH + c;
                float mold  = (t == 0) ? 0.0f : mem1[idx];
                float cur   = acc[tn][j] + bias;
                float reset = (mold > 1.0f) ? 1.0f : 0.0f;
                float mnew  = fmaf(bt1, mold, cur) - reset;
                mem1[idx]   = mnew;
                spk1[idx]   = (mnew > 1.0f) ? 1.0f : 0.0f;
            }
        }
    }
}

// ---------------------------------------------------------------------------
// Kernel B: one time step of layer 2 + output writes.
// grid = 128 blocks x 256 threads = 1024 waves; each wave does 4 rows.
// ---------------------------------------------------------------------------
__global__ __launch_bounds__(256) void snn_layer2_lif(
    const float* __restrict__ spk1,  // [B, NH]
    const float* __restrict__ W2,    // [NOUT, NH] row-major
    const float* __restrict__ b2,    // [NOUT]
    const float* __restrict__ beta2p,// [1]
    float* __restrict__ mem2,        // [B, NOUT] persistent state (ws)
    float* __restrict__ out_spk,     // [T, B, NOUT]
    float* __restrict__ out_mem,     // [T, B, NOUT]
    int t)
{
    __shared__ float w2s[NOUT * NH]; // 40 KB

    const int tid = threadIdx.x;
    // stage W2 (contiguous 10000 floats = 2500 float4)
    for (int i = tid; i < (NOUT * NH) / 4; i += 256)
        *(float4*)(&w2s[i * 4]) = *(const float4*)(W2 + i * 4);
    __syncthreads();

    const float bt2  = fminf(fmaxf(beta2p[0], 0.0f), 1.0f);
    const int lane   = tid & 31;
    const int wave   = tid >> 5;
    const int wgid   = blockIdx.x * 8 + wave;  // 0..1023

    for (int rr = 0; rr < 4; ++rr) {
        const int r = wgid * 4 + rr;           // 0..4095
        const float* srow = spk1 + (size_t)r * NH;

        float acc[NOUT];
#pragma unroll
        for (int o = 0; o < NOUT; ++o) acc[o] = 0.0f;

        // coalesced spk1 reads; per-lane LDS reads conflict-free
        for (int k = lane; k < NH; k += 32) {
            float s = srow[k];
#pragma unroll
            for (int o = 0; o < NOUT; ++o)
                acc[o] = fmaf(s, w2s[o * NH + k], acc[o]);
        }
        // wave32 xor-shuffle reduction
#pragma unroll
        for (int o = 0; o < NOUT; ++o) {
            float v = acc[o];
            v += __shfl_xor(v, 16, 32);
            v += __shfl_xor(v,  8, 32);
            v += __shfl_xor(v,  4, 32);
            v += __shfl_xor(v,  2, 32);
            v += __shfl_xor(v,  1, 32);
            acc[o] = v;
        }
        if (lane == 0) {
#pragma unroll
            for (int o = 0; o < NOUT; ++o) {
                size_t mi   = (size_t)r * NOUT + o;
                float mold  = (t == 0) ? 0.0f : mem2[mi];
                float cur   = acc[o] + b2[o];
                float reset = (mold > 1.0f) ? 1.0f : 0.0f;
                float mnew  = fmaf(bt2, mold, cur) - reset;
                mem2[mi] = mnew;
                size_t oi = (size_t)t * (BATCH * NOUT) + mi;
                out_spk[oi] = (mnew > 1.0f) ? 1.0f : 0.0f;
                out_mem[oi] = mnew;
            }
        }
    }
}

// ---------------------------------------------------------------------------
extern "C" void kernel_launch(void* const* d_in, const int* in_sizes, int n_in,
                              void* d_out, int out_size, void* d_ws, size_t ws_size,
                              hipStream_t stream)
{
    const float* x     = (const float*)d_in[0];
    const float* W1    = (const float*)d_in[1];
    const float* b1    = (const float*)d_in[2];
    const float* W2    = (const float*)d_in[3];
    const float* b2    = (const float*)d_in[4];
    const float* beta1 = (const float*)d_in[5];
    const float* beta2 = (const float*)d_in[6];

    float* out     = (float*)d_out;
    float* out_spk = out;                                // [T,B,NOUT]
    float* out_mem = out + (size_t)T_STEPS * BATCH * NOUT;

    // workspace layout: mem1 | spk1 | mem2  (~32.9 MB)
    float* mem1 = (float*)d_ws;
    float* spk1 = mem1 + (size_t)BATCH * NH;
    float* mem2 = spk1 + (size_t)BATCH * NH;

    dim3 block(256);
    dim3 gridA((NH + 127) / 128, BATCH / 128);   // (8, 32)
    dim3 gridB(128);

    for (int t = 0; t < T_STEPS; ++t) {
        snn_layer1_wmma<<<gridA, block, 0, stream>>>(x, W1, b1, beta1, mem1, spk1, t);
        snn_layer2_lif<<<gridB, block, 0, stream>>>(spk1, W2, b2, beta2, mem2,
                                                    out_spk, out_mem, t);
    }
}